// MyModel_61933428416721
// MI455X (gfx1250) — compile-verified
//
#include <hip/hip_runtime.h>

// Elementwise batched 2x2 inverse of H = [[1.2, cos(t)],[cos(t), 0.01]].
//   c   = cos(theta)
//   det = 0.012 - c*c
//   r   = 1/det
//   out = { 0.01*r, -c*r, -c*r, 1.2*r }   (16 contiguous bytes per element)
//
// Store-bandwidth-bound: 32 MB in + 128 MB out -> ~6.9 us floor at 23.3 TB/s.
// One dense global_store_b128 per lane, non-temporal hints on the streaming
// load/store, native v_cos_f32 / v_rcp_f32, global_prefetch_b8 for next tile.

// Native Clang vector type: accepted by __builtin_nontemporal_store
// (HIP's float4 class is not), same 16-byte layout.
typedef __attribute__((ext_vector_type(4))) float v4f;

__global__ __launch_bounds__(256) void inv2x2_stream_kernel(
    const float* __restrict__ theta, v4f* __restrict__ out, int n) {
  const int stride = (int)(blockDim.x * gridDim.x);
  for (int i = (int)(blockIdx.x * blockDim.x + threadIdx.x); i < n; i += stride) {
    // Speculative prefetch of the next grid-stride tile (gfx1250
    // global_prefetch_b8; silently dropped if OOB).
    __builtin_prefetch(theta + i + stride, 0, 0);

    // Read-once input: non-temporal load (TH=NT).
    float t = __builtin_nontemporal_load(theta + i);

    // Fast native path: v_mul (1/2pi scale) + v_cos_f32.
    float c = __cosf(t);

    // det = 1.2*0.01 - c*c, fused.
    float det = fmaf(-c, c, 0.012f);

    // v_rcp_f32 (~1 ulp); result only feeds 3 multiplies.
    float r = __builtin_amdgcn_rcpf(det);

    v4f v;
    v.x = 0.01f * r;   // d * inv_det
    v.y = -c * r;      // -c * inv_det
    v.z = v.y;         // symmetric off-diagonal
    v.w = 1.2f * r;    // a * inv_det

    // Write-once output: dense non-temporal global_store_b128.
    __builtin_nontemporal_store(v, out + i);
  }
}

extern "C" void kernel_launch(void* const* d_in, const int* in_sizes, int n_in,
                              void* d_out, int out_size, void* d_ws, size_t ws_size,
                              hipStream_t stream) {
  const float* theta = (const float*)d_in[0];
  v4f* out = (v4f*)d_out;                 // [B,2,2] f32 == B 16-byte vectors
  const int n = in_sizes[0];

  const int block = 256;                  // 8 waves (wave32) per workgroup
  // ~4 grid-stride iterations per thread: keeps the prefetch pipeline busy
  // while still giving every WGP a deep pool of waves to hide HBM latency.
  int grid = (n / 4 + block - 1) / block;
  if (grid < 1) grid = 1;

  inv2x2_stream_kernel<<<grid, block, 0, stream>>>(theta, out, n);
}